// GATConv_54296976556158
// MI455X (gfx1250) — compile-verified
//
#include <hip/hip_runtime.h>
#include <math.h>

#define N_NODES 100000
#define N_EDGES 1600000
#define IN_CH   128
#define HEADS   4
#define OUT_CH  32
#define HO      128          // HEADS*OUT_CH
#define NEG_SLOPE 0.2f
#define SCAN_B  1024
#define N_SCAN_BLOCKS ((N_NODES + SCAN_B - 1) / SCAN_B)   // 98

typedef __attribute__((ext_vector_type(2))) float v2f;
typedef __attribute__((ext_vector_type(8))) float v8f;

// ---------------------------------------------------------------------------
// K0: zero int buffer
// ---------------------------------------------------------------------------
__global__ void gat_zero_i32(int* __restrict__ p, int n) {
    int i = blockIdx.x * blockDim.x + threadIdx.x;
    if (i < n) p[i] = 0;
}

// ---------------------------------------------------------------------------
// K1: h = x @ W  via V_WMMA_F32_16X16X4_F32 (fp32, matches reference exactly
// in precision up to accumulation order). One block = 16 rows x 128 cols;
// 8 waves, each wave owns a 16x16 output tile, K-loop of 32 WMMAs.
// N_NODES = 6250 * 16 exactly, so no tail handling needed.
// ---------------------------------------------------------------------------
__global__ __launch_bounds__(256) void gat_gemm_wmma(
        const float* __restrict__ x, const float* __restrict__ w,
        float* __restrict__ h) {
    const int m0   = blockIdx.x * 16;
    const int wv   = threadIdx.x >> 5;          // 0..7 -> 16-col slice
    const int lane = threadIdx.x & 31;
    const int col  = (wv << 4) + (lane & 15);   // global output column
    const int row  = m0 + (lane & 15);          // A row for this lane
    const int kk   = (lane >> 4) << 1;          // lanes 0-15: K+0/1, 16-31: K+2/3

    const float* __restrict__ xr = x + (size_t)row * IN_CH;
    v8f c = {};
    for (int k = 0; k < IN_CH; k += 4) {
        v2f a, b;
        a.x = xr[k + kk];
        a.y = xr[k + kk + 1];
        b.x = w[(k + kk)     * HO + col];
        b.y = w[(k + kk + 1) * HO + col];
        c = __builtin_amdgcn_wmma_f32_16x16x4_f32(
                /*neg_a=*/false, a, /*neg_b=*/false, b,
                /*c_mod=*/(short)0, c, /*reuse_a=*/false, /*reuse_b=*/false);
    }
    // D layout: VGPR i -> lanes 0-15: M=i, lanes 16-31: M=i+8 ; N = lane&15 slice
    const int rbase = m0 + ((lane >> 4) << 3);
#pragma unroll
    for (int i = 0; i < 8; ++i)
        h[(size_t)(rbase + i) * HO + col] = c[i];
}

// ---------------------------------------------------------------------------
// K2: per-node attention scores  s_src[n][hd] = h[n,hd,:] . att_src[hd]
//                                s_dst[n][hd] = h[n,hd,:] . att_dst[hd]
// One wave per node, lane = channel, wave32 butterfly reduction.
// att layout: (HEADS, 2*OUT_CH): [hd*64 + 0..31]=src, [hd*64 + 32..63]=dst
// ---------------------------------------------------------------------------
__global__ __launch_bounds__(256) void gat_node_scores(
        const float* __restrict__ h, const float* __restrict__ att,
        float* __restrict__ s_src, float* __restrict__ s_dst) {
    const int node = (blockIdx.x * blockDim.x + threadIdx.x) >> 5;
    const int lane = threadIdx.x & 31;
    if (node >= N_NODES) return;
    const float* __restrict__ hr = h + (size_t)node * HO;
#pragma unroll
    for (int hd = 0; hd < HEADS; ++hd) {
        const float v  = hr[hd * OUT_CH + lane];
        float ps = v * att[hd * 2 * OUT_CH + lane];
        float pd = v * att[hd * 2 * OUT_CH + OUT_CH + lane];
#pragma unroll
        for (int off = 16; off > 0; off >>= 1) {
            ps += __shfl_xor(ps, off, 32);
            pd += __shfl_xor(pd, off, 32);
        }
        if (lane == 0) {
            s_src[node * HEADS + hd] = ps;
            s_dst[node * HEADS + hd] = pd;
        }
    }
}

// ---------------------------------------------------------------------------
// K3: in-degree histogram (int atomics, cheap: 1.6M ops)
// edge_index: int64, shape (2, E): row 0 = src, row 1 = dst
// ---------------------------------------------------------------------------
__global__ void gat_degree(const long long* __restrict__ ei, int* __restrict__ deg) {
    int e = blockIdx.x * blockDim.x + threadIdx.x;
    if (e >= N_EDGES) return;
    atomicAdd(&deg[(int)ei[N_EDGES + e]], 1);
}

// ---------------------------------------------------------------------------
// K4a/b/c: inclusive scan of deg -> incl ; cursor = incl - deg (row starts)
// ---------------------------------------------------------------------------
__global__ __launch_bounds__(SCAN_B) void gat_scan_block(
        const int* __restrict__ deg, int* __restrict__ incl, int* __restrict__ bsum) {
    __shared__ int sm[SCAN_B];
    const int gid = blockIdx.x * SCAN_B + threadIdx.x;
    sm[threadIdx.x] = (gid < N_NODES) ? deg[gid] : 0;
    __syncthreads();
    for (int off = 1; off < SCAN_B; off <<= 1) {
        int t = (threadIdx.x >= off) ? sm[threadIdx.x - off] : 0;
        __syncthreads();
        sm[threadIdx.x] += t;
        __syncthreads();
    }
    if (gid < N_NODES) incl[gid] = sm[threadIdx.x];
    if (threadIdx.x == SCAN_B - 1) bsum[blockIdx.x] = sm[SCAN_B - 1];
}

__global__ __launch_bounds__(128) void gat_scan_top(int* __restrict__ bsum, int nb) {
    __shared__ int sm[128];
    sm[threadIdx.x] = (threadIdx.x < nb) ? bsum[threadIdx.x] : 0;
    __syncthreads();
    for (int off = 1; off < 128; off <<= 1) {
        int t = (threadIdx.x >= off) ? sm[threadIdx.x - off] : 0;
        __syncthreads();
        sm[threadIdx.x] += t;
        __syncthreads();
    }
    if (threadIdx.x < nb) bsum[threadIdx.x] = sm[threadIdx.x];
}

__global__ __launch_bounds__(SCAN_B) void gat_scan_finish(
        int* __restrict__ incl, const int* __restrict__ bsum,
        const int* __restrict__ deg, int* __restrict__ cursor) {
    const int gid = blockIdx.x * SCAN_B + threadIdx.x;
    if (gid >= N_NODES) return;
    const int add = (blockIdx.x > 0) ? bsum[blockIdx.x - 1] : 0;
    const int v = incl[gid] + add;
    incl[gid]   = v;
    cursor[gid] = v - deg[gid];      // exclusive start
}

// ---------------------------------------------------------------------------
// K5: CSR fill (src ids grouped by dst)
// ---------------------------------------------------------------------------
__global__ void gat_csr_fill(const long long* __restrict__ ei,
                             int* __restrict__ cursor, int* __restrict__ csr_src) {
    int e = blockIdx.x * blockDim.x + threadIdx.x;
    if (e >= N_EDGES) return;
    const int s = (int)ei[e];
    const int d = (int)ei[N_EDGES + e];
    const int pos = atomicAdd(&cursor[d], 1);
    csr_src[pos] = s;
}

// ---------------------------------------------------------------------------
// K6: wave-per-destination softmax aggregation. No float atomics.
// Pass 1: lane-parallel max of logits per head; Pass 2: serial edge loop,
// lanes = 32 channels, 4 head accumulators per lane. All gathers L2-resident.
// ---------------------------------------------------------------------------
__global__ __launch_bounds__(256) void gat_aggregate(
        const float* __restrict__ h, const float* __restrict__ s_src,
        const float* __restrict__ s_dst, const int* __restrict__ incl,
        const int* __restrict__ csr_src, const float* __restrict__ bias,
        float* __restrict__ out) {
    const int dst  = (blockIdx.x * blockDim.x + threadIdx.x) >> 5;
    const int lane = threadIdx.x & 31;
    if (dst >= N_NODES) return;

    const int beg = dst ? incl[dst - 1] : 0;
    const int end = incl[dst];
    const float4 sd = *(const float4*)(s_dst + dst * HEADS);

    // Pass 1: per-head max logit (lane-parallel over edges, then butterfly)
    float m[HEADS] = {-1e30f, -1e30f, -1e30f, -1e30f};
    for (int e = beg + lane; e < end; e += 32) {
        const int s = csr_src[e];
        const float4 ss = *(const float4*)(s_src + s * HEADS);
        float a0 = ss.x + sd.x; a0 = (a0 >= 0.f) ? a0 : NEG_SLOPE * a0;
        float a1 = ss.y + sd.y; a1 = (a1 >= 0.f) ? a1 : NEG_SLOPE * a1;
        float a2 = ss.z + sd.z; a2 = (a2 >= 0.f) ? a2 : NEG_SLOPE * a2;
        float a3 = ss.w + sd.w; a3 = (a3 >= 0.f) ? a3 : NEG_SLOPE * a3;
        m[0] = fmaxf(m[0], a0); m[1] = fmaxf(m[1], a1);
        m[2] = fmaxf(m[2], a2); m[3] = fmaxf(m[3], a3);
    }
#pragma unroll
    for (int off = 16; off > 0; off >>= 1) {
        m[0] = fmaxf(m[0], __shfl_xor(m[0], off, 32));
        m[1] = fmaxf(m[1], __shfl_xor(m[1], off, 32));
        m[2] = fmaxf(m[2], __shfl_xor(m[2], off, 32));
        m[3] = fmaxf(m[3], __shfl_xor(m[3], off, 32));
    }

    // Pass 2: weighted accumulate; lane = channel, 4 heads per lane
    float acc[HEADS] = {0.f, 0.f, 0.f, 0.f};
    float sum[HEADS] = {0.f, 0.f, 0.f, 0.f};
    for (int e = beg; e < end; ++e) {
        const int s = csr_src[e];
        if (e + 1 < end)                               // hide L2 latency
            __builtin_prefetch(h + (size_t)csr_src[e + 1] * HO, 0, 0);
        const float4 ss = *(const float4*)(s_src + s * HEADS);
        float a0 = ss.x + sd.x; a0 = (a0 >= 0.f) ? a0 : NEG_SLOPE * a0;
        float a1 = ss.y + sd.y; a1 = (a1 >= 0.f) ? a1 : NEG_SLOPE * a1;
        float a2 = ss.z + sd.z; a2 = (a2 >= 0.f) ? a2 : NEG_SLOPE * a2;
        float a3 = ss.w + sd.w; a3 = (a3 >= 0.f) ? a3 : NEG_SLOPE * a3;
        const float w0 = __expf(a0 - m[0]);
        const float w1 = __expf(a1 - m[1]);
        const float w2 = __expf(a2 - m[2]);
        const float w3 = __expf(a3 - m[3]);
        const float* __restrict__ hr = h + (size_t)s * HO;
        acc[0] += w0 * hr[0 * OUT_CH + lane];  sum[0] += w0;
        acc[1] += w1 * hr[1 * OUT_CH + lane];  sum[1] += w1;
        acc[2] += w2 * hr[2 * OUT_CH + lane];  sum[2] += w2;
        acc[3] += w3 * hr[3 * OUT_CH + lane];  sum[3] += w3;
    }
#pragma unroll
    for (int hd = 0; hd < HEADS; ++hd)
        out[(size_t)dst * HO + hd * OUT_CH + lane] =
            acc[hd] / (sum[hd] + 1e-16f) + bias[hd * OUT_CH + lane];
}

// ---------------------------------------------------------------------------
extern "C" void kernel_launch(void* const* d_in, const int* in_sizes, int n_in,
                              void* d_out, int out_size, void* d_ws, size_t ws_size,
                              hipStream_t stream) {
    const float*     x    = (const float*)d_in[0];
    const long long* ei   = (const long long*)d_in[1];   // int64 (2, E)
    const float*     w    = (const float*)d_in[2];
    const float*     att  = (const float*)d_in[3];
    const float*     bias = (const float*)d_in[4];
    float*           out  = (float*)d_out;

    // carve scratch (~62 MB total, 256B-aligned slices)
    char* ws = (char*)d_ws;
    size_t off = 0;
    auto carve = [&](size_t bytes) -> char* {
        char* p = ws + off;
        off = (off + bytes + 255) & ~(size_t)255;
        return p;
    };
    float* h      = (float*)carve(sizeof(float) * (size_t)N_NODES * HO);
    float* s_src  = (float*)carve(sizeof(float) * (size_t)N_NODES * HEADS);
    float* s_dst  = (float*)carve(sizeof(float) * (size_t)N_NODES * HEADS);
    int*   deg    = (int*)  carve(sizeof(int)   * N_NODES);
    int*   incl   = (int*)  carve(sizeof(int)   * N_NODES);
    int*   cursor = (int*)  carve(sizeof(int)   * N_NODES);
    int*   bsum   = (int*)  carve(sizeof(int)   * 256);
    int*   csr    = (int*)  carve(sizeof(int)   * N_EDGES);

    gat_zero_i32 <<<(N_NODES + 255) / 256, 256, 0, stream>>>(deg, N_NODES);
    gat_gemm_wmma<<<N_NODES / 16, 256, 0, stream>>>(x, w, h);
    gat_node_scores<<<((size_t)N_NODES * 32 + 255) / 256, 256, 0, stream>>>(h, att, s_src, s_dst);
    gat_degree   <<<(N_EDGES + 255) / 256, 256, 0, stream>>>(ei, deg);
    gat_scan_block<<<N_SCAN_BLOCKS, SCAN_B, 0, stream>>>(deg, incl, bsum);
    gat_scan_top  <<<1, 128, 0, stream>>>(bsum, N_SCAN_BLOCKS);
    gat_scan_finish<<<N_SCAN_BLOCKS, SCAN_B, 0, stream>>>(incl, bsum, deg, cursor);
    gat_csr_fill  <<<(N_EDGES + 255) / 256, 256, 0, stream>>>(ei, cursor, csr);
    gat_aggregate <<<((size_t)N_NODES * 32 + 255) / 256, 256, 0, stream>>>(
        h, s_src, s_dst, incl, csr, bias, out);
}